// DBLoss_44762149159355
// MI455X (gfx1250) — compile-verified
//
#include <hip/hip_runtime.h>
#include <stdint.h>

#define NI  16
#define HH  640
#define WW  640
#define HWP (HH*WW)          /* 409600 pixels per image  */
#define NHW (NI*HWP)         /* 6553600 total pixels     */
#define EPSP 1e-7f

#define LOSS_BLOCKS 1280     /* tiles: 6400 = 1280 * 5   */
#define TILES_PER_IMG 400    /* 409600 / 1024            */
#define NUM_TILES 6400

typedef float v2f __attribute__((ext_vector_type(2)));
typedef float v8f __attribute__((ext_vector_type(8)));

// ---------- orderable float <-> key (monotone, bijective for non-NaN) ----------
__device__ __forceinline__ uint32_t f2key(float f) {
  uint32_t b = __float_as_uint(f);
  return (b & 0x80000000u) ? ~b : (b | 0x80000000u);
}
__device__ __forceinline__ float key2f(uint32_t k) {
  return (k & 0x80000000u) ? __uint_as_float(k & 0x7fffffffu) : __uint_as_float(~k);
}

// ---------- CDNA5 async global->LDS (ASYNCcnt-tracked DMA path) ----------
__device__ __forceinline__ void async_ld_b128(unsigned lds_addr, unsigned voff_bytes,
                                              unsigned long long base) {
  asm volatile("global_load_async_to_lds_b128 %0, %1, %2"
               :: "v"(lds_addr), "v"(voff_bytes), "s"(base) : "memory");
}
__device__ __forceinline__ void wait_async0() { asm volatile("s_wait_asynccnt 0" ::: "memory"); }
__device__ __forceinline__ void wait_async6() { asm volatile("s_wait_asynccnt 6" ::: "memory"); }

// =====================================================================
// 0) zero histogram + per-image counters
// =====================================================================
__global__ void k_init(uint32_t* w, int nwords) {
  int i = blockIdx.x * blockDim.x + threadIdx.x;
  int stride = gridDim.x * blockDim.x;
  for (; i < nwords; i += stride) w[i] = 0u;
}

// =====================================================================
// 1) pass A: per-image pos/neg counts + 65536-bin histogram of key>>16
// =====================================================================
__global__ __launch_bounds__(256) void k_passA(const float* __restrict__ outs,
                                               const float* __restrict__ labs,
                                               const float* __restrict__ tmv,
                                               uint32_t* __restrict__ ghist,
                                               uint32_t* __restrict__ posn,
                                               uint32_t* __restrict__ nega) {
  __shared__ uint32_t sp[NI], sn[NI];
  if (threadIdx.x < NI) { sp[threadIdx.x] = 0u; sn[threadIdx.x] = 0u; }
  __syncthreads();
  int stride = gridDim.x * blockDim.x;
  for (int idx = blockIdx.x * blockDim.x + threadIdx.x; idx < NHW; idx += stride) {
    unsigned n  = (unsigned)idx / HWP;          // uniform per wave (HWP % 32 == 0)
    unsigned hw = (unsigned)idx - n * HWP;
    float p = outs[(size_t)(2u * n) * HWP + hw];
    float g = labs[(size_t)(2u * n) * HWP + hw];
    float m = tmv[idx];
    bool pos = (g >  0.5f) && (m > 0.5f);
    bool neg = (g <= 0.5f) && (m > 0.5f);
    if (neg) atomicAdd(&ghist[(n << 16) + (f2key(p) >> 16)], 1u);
    unsigned long long bp = __ballot(pos);
    unsigned long long bn = __ballot(neg);
    if ((threadIdx.x & 31) == 0) {
      if (bp) atomicAdd(&sp[n], (uint32_t)__popcll(bp));
      if (bn) atomicAdd(&sn[n], (uint32_t)__popcll(bn));
    }
  }
  __syncthreads();
  if (threadIdx.x < NI) {
    if (sp[threadIdx.x]) atomicAdd(&posn[threadIdx.x], sp[threadIdx.x]);
    if (sn[threadIdx.x]) atomicAdd(&nega[threadIdx.x], sn[threadIdx.x]);
  }
}

// =====================================================================
// 2) select A: find high-16-bit prefix of k-th largest; re-zero hist
// =====================================================================
__global__ __launch_bounds__(256) void k_selectA(uint32_t* __restrict__ ghist,
                                                 const uint32_t* __restrict__ posn,
                                                 const uint32_t* __restrict__ nega,
                                                 uint32_t* __restrict__ karr,
                                                 uint32_t* __restrict__ pref,
                                                 uint32_t* __restrict__ remv) {
  int n = blockIdx.x;
  __shared__ uint32_t csum[256];
  __shared__ uint32_t sK, sChunk, sKin;
  uint32_t* h = ghist + ((unsigned)n << 16);
  if (threadIdx.x == 0) {
    uint32_t p = posn[n], a = nega[n];
    uint32_t k = 3u * p; if (k > a) k = a;
    karr[n] = k; sK = k; sChunk = 0xFFFFFFFFu;
  }
  int b0 = threadIdx.x << 8;
  uint32_t s = 0;
  for (int j = 0; j < 256; ++j) s += h[b0 + j];
  csum[threadIdx.x] = s;
  __syncthreads();
  if (threadIdx.x == 0) {
    uint32_t k = sK;
    if (k) {
      uint32_t cum = 0;
      for (int t = 255; t >= 0; --t) {
        if (cum + csum[t] >= k) { sChunk = (uint32_t)t; sKin = k - cum; break; }
        cum += csum[t];
      }
    }
  }
  __syncthreads();
  if (sChunk == 0xFFFFFFFFu) {
    if (threadIdx.x == 0) { pref[n] = 0xFFFFFFFFu; remv[n] = 0u; }
  } else if (threadIdx.x == (int)sChunk) {
    uint32_t kin = sKin, c = 0;
    for (int j = 255; j >= 0; --j) {
      uint32_t hh = h[b0 + j];
      if (c + hh >= kin) { pref[n] = (uint32_t)(b0 + j); remv[n] = kin - c; break; }
      c += hh;
    }
  }
  __syncthreads();
  for (int j = 0; j < 256; ++j) h[b0 + j] = 0u;   // ready for pass B
}

// =====================================================================
// 3) pass B: low-16-bit histogram of keys matching prefix
// =====================================================================
__global__ __launch_bounds__(256) void k_passB(const float* __restrict__ outs,
                                               const float* __restrict__ labs,
                                               const float* __restrict__ tmv,
                                               uint32_t* __restrict__ ghist,
                                               const uint32_t* __restrict__ pref) {
  int stride = gridDim.x * blockDim.x;
  for (int idx = blockIdx.x * blockDim.x + threadIdx.x; idx < NHW; idx += stride) {
    unsigned n  = (unsigned)idx / HWP;
    unsigned hw = (unsigned)idx - n * HWP;
    uint32_t pf = pref[n];
    float p = outs[(size_t)(2u * n) * HWP + hw];
    float g = labs[(size_t)(2u * n) * HWP + hw];
    float m = tmv[idx];
    bool neg = (g <= 0.5f) && (m > 0.5f);
    uint32_t key = f2key(p);
    if (neg && (pf <= 0xFFFFu) && ((key >> 16) == pf))
      atomicAdd(&ghist[(n << 16) + (key & 0xFFFFu)], 1u);
  }
}

// =====================================================================
// 4) select B: exact threshold value per image
// =====================================================================
__global__ __launch_bounds__(256) void k_selectB(const uint32_t* __restrict__ ghist,
                                                 const uint32_t* __restrict__ karr,
                                                 const uint32_t* __restrict__ pref,
                                                 const uint32_t* __restrict__ remv,
                                                 float* __restrict__ thr) {
  int n = blockIdx.x;
  __shared__ uint32_t csum[256];
  __shared__ uint32_t sChunk, sKin;
  const uint32_t* h = ghist + ((unsigned)n << 16);
  int b0 = threadIdx.x << 8;
  uint32_t s = 0;
  for (int j = 0; j < 256; ++j) s += h[b0 + j];
  csum[threadIdx.x] = s;
  if (threadIdx.x == 0) sChunk = 0xFFFFFFFFu;
  __syncthreads();
  if (threadIdx.x == 0) {
    uint32_t k = remv[n];
    if (karr[n] == 0u) {
      thr[n] = __uint_as_float(0x7F800000u);      // +inf -> sel degenerates to pos
    } else {
      uint32_t cum = 0;
      for (int t = 255; t >= 0; --t) {
        if (cum + csum[t] >= k) { sChunk = (uint32_t)t; sKin = k - cum; break; }
        cum += csum[t];
      }
    }
  }
  __syncthreads();
  if (sChunk != 0xFFFFFFFFu && threadIdx.x == (int)sChunk) {
    uint32_t kin = sKin, c = 0;
    for (int j = 255; j >= 0; --j) {
      uint32_t hh = h[b0 + j];
      if (c + hh >= kin) {
        thr[n] = key2f((pref[n] << 16) | (uint32_t)(b0 + j));
        break;
      }
      c += hh;
    }
  }
}

// =====================================================================
// 5) fused loss kernel: async global->LDS double-buffered staging of the
//    six streams; 5 deterministic block partial sums
// =====================================================================
__global__ __launch_bounds__(256) void k_loss(const float* __restrict__ outs,
                                              const float* __restrict__ labs,
                                              const float* __restrict__ tmv,
                                              const float* __restrict__ gdv,
                                              const float* __restrict__ thrv,
                                              const uint32_t* __restrict__ posn,
                                              const uint32_t* __restrict__ karr,
                                              float* __restrict__ part) {
  __shared__ __align__(16) float sm[2][6][1024];
  __shared__ float sred[8][5];
  const unsigned tid = threadIdx.x;
  const unsigned smBase = (unsigned)(size_t)(&sm[0][0][0]);

  auto issue = [&](int it, int buf) {
    unsigned tile = blockIdx.x + (unsigned)it * LOSS_BLOCKS;
    unsigned n    = tile / TILES_PER_IMG;
    unsigned hwb  = (tile - n * TILES_PER_IMG) * 1024u;
    unsigned pOff = (2u * n * HWP + hwb + tid * 4u) * 4u;   // bytes
    unsigned tOff = pOff + HWP * 4u;
    unsigned xOff = (n * HWP + hwb + tid * 4u) * 4u;
    unsigned dst  = smBase + (unsigned)buf * (6u * 4096u) + tid * 16u;
    async_ld_b128(dst + 0u * 4096u, pOff, (unsigned long long)outs);
    async_ld_b128(dst + 1u * 4096u, tOff, (unsigned long long)outs);
    async_ld_b128(dst + 2u * 4096u, pOff, (unsigned long long)labs);
    async_ld_b128(dst + 3u * 4096u, tOff, (unsigned long long)labs);
    async_ld_b128(dst + 4u * 4096u, xOff, (unsigned long long)tmv);
    async_ld_b128(dst + 5u * 4096u, xOff, (unsigned long long)gdv);
  };

  float aSel = 0.f, aLp = 0.f, aLb = 0.f, aTh = 0.f, aGd = 0.f;

  issue(0, 0);
  for (int it = 0; it < 5; ++it) {
    if (it + 1 < 5) { issue(it + 1, (it + 1) & 1); wait_async6(); }
    else            { wait_async0(); }
    int buf = it & 1;
    unsigned tile = blockIdx.x + (unsigned)it * LOSS_BLOCKS;
    unsigned n = tile / TILES_PER_IMG;
    float thr = thrv[n];
    uint32_t pn = posn[n], kk = karr[n];
#pragma unroll
    for (int j = 0; j < 4; ++j) {
      int e = (int)tid * 4 + j;
      float p  = sm[buf][0][e];
      float x  = sm[buf][1][e];
      float g  = sm[buf][2][e];
      float gt = sm[buf][3][e];
      float m  = sm[buf][4][e];
      float gw = sm[buf][5][e];
      bool pos = (g >  0.5f) && (m > 0.5f);
      bool neg = (g <= 0.5f) && (m > 0.5f);
      float sel;
      if (pn == 0u)       sel = (m > 0.5f) ? 1.f : 0.f;
      else if (kk == 0u)  sel = pos ? 1.f : 0.f;
      else                sel = ((neg && (p >= thr)) || pos) ? 1.f : 0.f;
      bool  t1 = (g > 0.5f);
      float pc = fminf(fmaxf(p, EPSP), 1.f - EPSP);
      float lp = t1 ? -logf(pc) : -log1pf(-pc);
      float xx = 50.f * (p - x);
      float bs = 1.f / (1.f + expf(-xx));
      float bc = fminf(fmaxf(bs, EPSP), 1.f - EPSP);
      float lb = t1 ? -logf(bc) : -log1pf(-bc);
      aSel += sel;
      aLp  += lp * sel;
      aLb  += lb * sel;
      aTh  += fabsf(x - gt) * gw;
      aGd  += gw;
    }
  }

  float v[5] = { aSel, aLp, aLb, aTh, aGd };
#pragma unroll
  for (int a = 0; a < 5; ++a)
    for (int off = 16; off > 0; off >>= 1) v[a] += __shfl_down(v[a], off, 32);
  int wave = tid >> 5, lane = tid & 31;
  if (lane == 0)
    for (int a = 0; a < 5; ++a) sred[wave][a] = v[a];
  __syncthreads();
  if (tid < 5) {
    float s = 0.f;
    for (int w = 0; w < 8; ++w) s += sred[w][tid];
    part[tid * LOSS_BLOCKS + blockIdx.x] = s;
  }
}

// =====================================================================
// 6) finalize: exact fp32 WMMA-chain reduction (D = A x ones + C sums
//    64 partials per instruction), one wave, deterministic
// =====================================================================
__global__ __launch_bounds__(32) void k_finalize(const float* __restrict__ part,
                                                 float* __restrict__ out) {
  const int L  = threadIdx.x;
  const int r  = L & 15;
  const int c0 = (L < 16) ? 0 : 2;
  v8f zero = { 0.f, 0.f, 0.f, 0.f, 0.f, 0.f, 0.f, 0.f };
  v8f acc[5] = { zero, zero, zero, zero, zero };
  v2f ones;  ones.x = 1.f; ones.y = 1.f;

  for (int itr = 0; itr < LOSS_BLOCKS / 64; ++itr) {
    int base = itr * 64;
#pragma unroll
    for (int a = 0; a < 5; ++a) {
      const float* src = part + a * LOSS_BLOCKS + base + 4 * r + c0;
      v2f av;  av.x = src[0];  av.y = src[1];
      acc[a] = __builtin_amdgcn_wmma_f32_16x16x4_f32(
          false, av, false, ones, (short)0, acc[a], false, false);
    }
  }
  float tot[5];
#pragma unroll
  for (int a = 0; a < 5; ++a) {
    float s = 0.f;
    for (int j = 0; j < 8; ++j) s += acc[a][j];
    tot[a] = __shfl(s, 0, 32) + __shfl(s, 16, 32);
  }
  if (L == 0) {
    float selsum = tot[0], lp = tot[1], lb = tot[2], th = tot[3], gd = tot[4];
    float lprob = (selsum > 0.f) ? (lp / selsum) : 0.f;
    float lbin  = (selsum > 0.f) ? (lb / selsum) : 0.f;
    float lth   = th / (gd + 1e-6f);
    out[0] = lprob + 1.0f * lbin + 10.0f * lth;
    out[1] = lprob;
    out[2] = lbin;
    out[3] = lth;
  }
}

// =====================================================================
extern "C" void kernel_launch(void* const* d_in, const int* in_sizes, int n_in,
                              void* d_out, int out_size, void* d_ws, size_t ws_size,
                              hipStream_t stream) {
  (void)in_sizes; (void)n_in; (void)out_size; (void)ws_size;
  const float* outs = (const float*)d_in[0];
  const float* labs = (const float*)d_in[1];
  const float* tmv  = (const float*)d_in[2];
  const float* gdv  = (const float*)d_in[3];

  uint32_t* ghist = (uint32_t*)d_ws;            // NI*65536 words (4 MB)
  uint32_t* posn  = ghist + (size_t)NI * 65536;
  uint32_t* nega  = posn + NI;
  uint32_t* karr  = nega + NI;
  uint32_t* pref  = karr + NI;
  uint32_t* remv  = pref + NI;
  float*    thr   = (float*)(remv + NI);
  float*    part  = thr + NI;                   // 5 * LOSS_BLOCKS floats

  int zwords = NI * 65536 + 6 * NI;
  k_init   <<<1024, 256, 0, stream>>>(ghist, zwords);
  k_passA  <<<1024, 256, 0, stream>>>(outs, labs, tmv, ghist, posn, nega);
  k_selectA<<<NI,   256, 0, stream>>>(ghist, posn, nega, karr, pref, remv);
  k_passB  <<<1024, 256, 0, stream>>>(outs, labs, tmv, ghist, pref);
  k_selectB<<<NI,   256, 0, stream>>>(ghist, karr, pref, remv, thr);
  k_loss   <<<LOSS_BLOCKS, 256, 0, stream>>>(outs, labs, tmv, gdv, thr, posn, karr, part);
  k_finalize<<<1, 32, 0, stream>>>(part, (float*)d_out);
}